// SPETimeEmbeddingModel_90735479095614
// MI455X (gfx1250) — compile-verified
//
#include <hip/hip_runtime.h>
#include <hip/hip_bf16.h>

typedef __attribute__((ext_vector_type(2))) float v2f;
typedef __attribute__((ext_vector_type(4))) float v4f;
typedef __attribute__((ext_vector_type(8))) float v8f;

#define BB 8192
#define SS 32
#define D_NODE 172
#define D_EDGE 172
#define D_TIME 100
#define D_IN 616          // 172 + 172 + 172 + 100
#define LDS_STRIDE 620    // 2480 B/row: 16B-aligned rows; 620%64=44 -> 16 distinct banks over M=0..15

__global__ __launch_bounds__(256) void tgnn_fused_kernel(
    const float* __restrict__ node_feats,   // [200000,172]
    const float* __restrict__ edge_feats,   // [500000,172]
    const int*   __restrict__ src_ids,      // [B]
    const int*   __restrict__ dst_ids,      // [B]
    const float* __restrict__ itimes,       // [B]
    const int*   __restrict__ src_nbr,  const int* __restrict__ src_edge, const float* __restrict__ src_nt,
    const int*   __restrict__ dst_nbr,  const int* __restrict__ dst_edge, const float* __restrict__ dst_nt,
    const float* __restrict__ time_w,   const float* __restrict__ time_b,
    const float* __restrict__ Wout,     // [616,172]
    const float* __restrict__ bout,     // [172]
    float* __restrict__ out)            // [3, B, 172] flat
{
    __shared__ float inp[16][LDS_STRIDE];   // assembled [16 x 616] input tile
    __shared__ int   s_nbr[16][SS];
    __shared__ int   s_edg[16][SS];
    __shared__ float s_dt[16][SS];
    __shared__ float s_val[16][SS];

    const int tid  = threadIdx.x;
    const int side = blockIdx.y;            // 0 = src, 1 = dst
    const int row0 = blockIdx.x * 16;

    const int*   ids = side ? dst_ids  : src_ids;
    const int*   nbr = side ? dst_nbr  : src_nbr;
    const int*   edg = side ? dst_edge : src_edge;
    const float* ntm = side ? dst_nt   : src_nt;

    // ---- stage metadata for the 16 rows x 32 neighbors ----
    for (int idx = tid; idx < 16 * SS; idx += 256) {
        int rr = idx >> 5, ss = idx & 31;
        int grow = row0 + rr;
        int nid  = nbr[grow * SS + ss];
        s_nbr[rr][ss] = nid;
        s_edg[rr][ss] = edg[grow * SS + ss];
        s_dt[rr][ss]  = itimes[grow] - ntm[grow * SS + ss];
        s_val[rr][ss] = (nid == 0) ? 0.0f : 1.0f;   // zero padded time features
    }
    __syncthreads();

    // ---- phase 1: neighbor gather (b128) + time encode + mean over S ----
    const int r   = tid >> 4;   // batch row within tile (0..15)
    const int c16 = tid & 15;   // 16 lanes cover the 43 float4 chunks of a 172-f32 row

    v4f aN0 = {0,0,0,0}, aN1 = {0,0,0,0}, aN2 = {0,0,0,0};
    v4f aE0 = {0,0,0,0}, aE1 = {0,0,0,0}, aE2 = {0,0,0,0};
    float accT[7], tw[7], tb[7];
    #pragma unroll
    for (int j = 0; j < 7; ++j) {
        int d = c16 + 16 * j;
        accT[j] = 0.f;
        tw[j] = (d < D_TIME) ? time_w[d] : 0.f;
        tb[j] = (d < D_TIME) ? time_b[d] : 0.f;
    }
    const bool tail = (c16 < 11);   // chunk c16+32 < 43

    for (int s = 0; s < SS; ++s) {
        const v4f* nrow = (const v4f*)(node_feats + (size_t)s_nbr[r][s] * D_NODE);
        const v4f* erow = (const v4f*)(edge_feats + (size_t)s_edg[r][s] * D_EDGE);
        aN0 += nrow[c16];
        aN1 += nrow[c16 + 16];
        aE0 += erow[c16];
        aE1 += erow[c16 + 16];
        if (tail) {
            aN2 += nrow[c16 + 32];
            aE2 += erow[c16 + 32];
        }
        float dt = s_dt[r][s];
        float vm = s_val[r][s];
        #pragma unroll
        for (int j = 0; j < 6; ++j)
            accT[j] += vm * __cosf(dt * tw[j] + tb[j]);
        if (c16 < 4)
            accT[6] += vm * __cosf(dt * tw[6] + tb[6]);
    }

    const float inv = 1.0f / (float)SS;
    // neighbor-node mean at [172,344), edge mean at [344,516)  (all 16B-aligned in LDS)
    *(v4f*)&inp[r][D_NODE + 4 * c16]           = aN0 * inv;
    *(v4f*)&inp[r][D_NODE + 64 + 4 * c16]      = aN1 * inv;
    *(v4f*)&inp[r][2 * D_NODE + 4 * c16]       = aE0 * inv;
    *(v4f*)&inp[r][2 * D_NODE + 64 + 4 * c16]  = aE1 * inv;
    if (tail) {
        *(v4f*)&inp[r][D_NODE + 128 + 4 * c16]     = aN2 * inv;
        *(v4f*)&inp[r][2 * D_NODE + 128 + 4 * c16] = aE2 * inv;
    }
    // time mean at [516,616)
    #pragma unroll
    for (int j = 0; j < 7; ++j) {
        int d = c16 + 16 * j;
        if (d < D_TIME)
            inp[r][2 * D_NODE + D_EDGE + d] = accT[j] * inv;
    }
    {   // center node features at [0,172)
        const v4f* crow = (const v4f*)(node_feats + (size_t)ids[row0 + r] * D_NODE);
        *(v4f*)&inp[r][4 * c16]      = crow[c16];
        *(v4f*)&inp[r][64 + 4 * c16] = crow[c16 + 16];
        if (tail) *(v4f*)&inp[r][128 + 4 * c16] = crow[c16 + 32];
    }
    __syncthreads();

    // ---- phase 2: [16x616] @ [616x172] via v_wmma_f32_16x16x4_f32, relu ----
    const int  w    = tid >> 5;        // wave id (0..7), each owns N-tiles w, w+8
    const int  lane = tid & 31;
    const int  nl   = lane & 15;
    const bool hi   = lane >= 16;
    const size_t outBase = (size_t)side * BB * D_NODE;

    for (int nt = w; nt < 11; nt += 8) {
        const int n0   = nt * 16;
        const int ncol = n0 + nl;
        const int ncl  = (ncol < D_NODE) ? ncol : (D_NODE - 1);  // clamp loads, guard stores
        v8f c = {0.f, 0.f, 0.f, 0.f, 0.f, 0.f, 0.f, 0.f};
        #pragma unroll 2
        for (int k0 = 0; k0 < D_IN; k0 += 4) {
            // A 16x4: lanes 0-15 hold (M=lane, K=k0,k0+1); lanes 16-31 hold (M=lane-16, K=k0+2,k0+3)
            const int ka = k0 + (hi ? 2 : 0);
            v2f a = *(const v2f*)&inp[nl][ka];
            // B 4x16: vgpr0 = rows K=k0 (lo lanes) / K=k0+2 (hi lanes); vgpr1 = K=k0+1 / K=k0+3
            v2f b;
            b.x = Wout[(size_t)ka       * D_NODE + ncl];
            b.y = Wout[(size_t)(ka + 1) * D_NODE + ncl];
            c = __builtin_amdgcn_wmma_f32_16x16x4_f32(
                    false, a, false, b, (short)0, c, false, false);
        }
        if (ncol < D_NODE) {
            const float bias  = bout[ncol];
            const int   mbase = hi ? 8 : 0;   // C/D: vgpr i -> M=i (lo lanes) / M=i+8 (hi lanes)
            #pragma unroll
            for (int i = 0; i < 8; ++i) {
                float v = c[i] + bias;
                v = (v > 0.f) ? v : 0.f;
                out[outBase + (size_t)(row0 + mbase + i) * D_NODE + ncol] = v;
            }
        }
    }
}

__global__ void tgnn_zero_kernel(float* __restrict__ p, int n) {
    int i = blockIdx.x * blockDim.x + threadIdx.x;
    if (i < n) p[i] = 0.0f;
}

extern "C" void kernel_launch(void* const* d_in, const int* in_sizes, int n_in,
                              void* d_out, int out_size, void* d_ws, size_t ws_size,
                              hipStream_t stream) {
    (void)in_sizes; (void)n_in; (void)out_size; (void)d_ws; (void)ws_size;

    const float* node_feats = (const float*)d_in[0];
    const float* edge_feats = (const float*)d_in[1];
    const int*   src_ids    = (const int*)  d_in[2];
    const int*   dst_ids    = (const int*)  d_in[3];
    const float* itimes     = (const float*)d_in[4];
    const int*   src_nbr    = (const int*)  d_in[5];
    const int*   src_edge   = (const int*)  d_in[6];
    const float* src_nt     = (const float*)d_in[7];
    const int*   dst_nbr    = (const int*)  d_in[8];
    const int*   dst_edge   = (const int*)  d_in[9];
    const float* dst_nt     = (const float*)d_in[10];
    const float* time_w     = (const float*)d_in[11];
    const float* time_b     = (const float*)d_in[12];
    const float* Wout       = (const float*)d_in[13];
    const float* bout       = (const float*)d_in[14];
    float* out = (float*)d_out;

    dim3 grid(BB / 16, 2);   // 512 batch tiles x {src,dst}
    tgnn_fused_kernel<<<grid, 256, 0, stream>>>(
        node_feats, edge_feats, src_ids, dst_ids, itimes,
        src_nbr, src_edge, src_nt, dst_nbr, dst_edge, dst_nt,
        time_w, time_b, Wout, bout, out);

    const int ndummy = BB * D_NODE;  // third output: zeros_like(src_emb)
    tgnn_zero_kernel<<<(ndummy + 255) / 256, 256, 0, stream>>>(
        out + (size_t)2 * BB * D_NODE, ndummy);
}